// SpGraphAttentionLayer_60069412601882
// MI455X (gfx1250) — compile-verified
//
#include <hip/hip_runtime.h>
#include <math.h>

typedef float v2f __attribute__((ext_vector_type(2)));
typedef float v8f __attribute__((ext_vector_type(8)));

#define DIN   128
#define DOUT  128
#define ALPHA 0.2f
#define SCALEF 10.0f
#define EPSF  1e-15f

__device__ __forceinline__ float wave_reduce_add(float v) {
  #pragma unroll
  for (int off = 16; off > 0; off >>= 1) v += __shfl_down(v, off, 32);
  return v;
}

// ---------------- Kernel 0: zero accumulators ----------------
__global__ __launch_bounds__(256) void k_zero(float* __restrict__ out,
                                              float* __restrict__ rowsum,
                                              int n_nodes) {
  int tid = blockIdx.x * blockDim.x + threadIdx.x;
  int total4 = n_nodes * (DOUT / 4);
  if (tid < total4) ((float4*)out)[tid] = make_float4(0.f, 0.f, 0.f, 0.f);
  if (tid < n_nodes) rowsum[tid] = 0.f;
}

// ---------------- Kernel 1: h = x @ W  (fp32 WMMA 16x16x4) ----------------
// 1 block = one 16-row strip of x; 8 waves each own one 16-col tile of W.
__global__ __launch_bounds__(256) void k_gemm(const float* __restrict__ x,
                                              const float* __restrict__ W,
                                              float* __restrict__ h,
                                              int n_nodes) {
  __shared__ float ldsW[DIN * DOUT];           // 64 KB of the 320 KB WGP LDS
  for (int i = threadIdx.x; i < DIN * DOUT / 4; i += 256)
    ((float4*)ldsW)[i] = ((const float4*)W)[i];
  __syncthreads();

  const int wave  = threadIdx.x >> 5;
  const int lane  = threadIdx.x & 31;
  const int half  = lane >> 4;                 // 0: K lo pair, 1: K hi pair
  const int l16   = lane & 15;
  const int mbase = blockIdx.x * 16;
  const int nbase = wave * 16;

  int arow = mbase + l16;
  if (arow >= n_nodes) arow = n_nodes - 1;     // clamp (stores are guarded)
  const float* xrow = x + (size_t)arow * DIN;

  v8f acc = {};
  #pragma unroll
  for (int ks = 0; ks < DIN / 4; ++ks) {
    const int k = ks * 4 + 2 * half;
    v2f a; a.x = xrow[k];                a.y = xrow[k + 1];
    v2f b; b.x = ldsW[k * DOUT + nbase + l16];
           b.y = ldsW[(k + 1) * DOUT + nbase + l16];
    acc = __builtin_amdgcn_wmma_f32_16x16x4_f32(false, a, false, b,
                                                (short)0, acc, false, false);
  }

  #pragma unroll
  for (int r = 0; r < 8; ++r) {
    const int row = mbase + r + 8 * half;      // C layout: VGPR r, lane half
    if (row < n_nodes) h[(size_t)row * DOUT + nbase + l16] = acc[r];
  }
}

// ---------------- Kernel 2: per-node attention scalars ----------------
// s1[n] = h[n] . a[0:128], s2[n] = h[n] . a[128:256].  One wave per node.
__global__ __launch_bounds__(256) void k_s1s2(const float* __restrict__ h,
                                              const float* __restrict__ a,
                                              float* __restrict__ s1,
                                              float* __restrict__ s2,
                                              int n_nodes) {
  const int node = (blockIdx.x * blockDim.x + threadIdx.x) >> 5;
  const int lane = threadIdx.x & 31;
  if (node >= n_nodes) return;
  float4 hv = ((const float4*)(h + (size_t)node * DOUT))[lane];
  float4 a1 = ((const float4*)a)[lane];
  float4 a2 = ((const float4*)(a + DOUT))[lane];
  float d1 = hv.x * a1.x + hv.y * a1.y + hv.z * a1.z + hv.w * a1.w;
  float d2 = hv.x * a2.x + hv.y * a2.y + hv.z * a2.z + hv.w * a2.w;
  d1 = wave_reduce_add(d1);
  d2 = wave_reduce_add(d2);
  if (lane == 0) { s1[node] = d1; s2[node] = d2; }
}

// ---------------- Kernel 3: per-edge weighted scatter ----------------
// One wave per edge: e = exp(-leakyrelu(s1[src]+s2[dst])),
// out[src] += e * h[dst]  (fp32 atomics, L2-resident), rowsum[src] += e.
__global__ __launch_bounds__(256) void k_edge(const int* __restrict__ src,
                                              const int* __restrict__ dst,
                                              const float* __restrict__ h,
                                              const float* __restrict__ s1,
                                              const float* __restrict__ s2,
                                              float* __restrict__ acc,
                                              float* __restrict__ rowsum,
                                              int n_edges) {
  const int e    = (blockIdx.x * blockDim.x + threadIdx.x) >> 5;
  const int lane = threadIdx.x & 31;
  if (e >= n_edges) return;

  int s = 0, d = 0;
  float ev = 0.f;
  if (lane == 0) {
    s = src[e];
    d = dst[e];
    float logit = s1[s] + s2[d];
    float lr = logit > 0.f ? logit : ALPHA * logit;
    ev = expf(-lr);
    atomicAdd(rowsum + s, ev);
  }
  s  = __shfl(s, 0, 32);
  d  = __shfl(d, 0, 32);
  ev = __shfl(ev, 0, 32);

  float4 hd = ((const float4*)(h + (size_t)d * DOUT))[lane];
  float* o = acc + (size_t)s * DOUT + 4 * lane;
  atomicAdd(o + 0, ev * hd.x);
  atomicAdd(o + 1, ev * hd.y);
  atomicAdd(o + 2, ev * hd.z);
  atomicAdd(o + 3, ev * hd.w);
}

// ---------------- Kernel 4: normalize + ReLU + expmap0 ----------------
// One wave per node; in-place on d_out.
__global__ __launch_bounds__(256) void k_final(float* __restrict__ out,
                                               const float* __restrict__ rowsum,
                                               int n_nodes) {
  const int node = (blockIdx.x * blockDim.x + threadIdx.x) >> 5;
  const int lane = threadIdx.x & 31;
  if (node >= n_nodes) return;

  const float inv = 1.f / fmaxf(rowsum[node], EPSF);
  float4 v = ((float4*)(out + (size_t)node * DOUT))[lane];
  float4 u;
  u.x = fmaxf(v.x * inv, 0.f) * (1.f / SCALEF);
  u.y = fmaxf(v.y * inv, 0.f) * (1.f / SCALEF);
  u.z = fmaxf(v.z * inv, 0.f) * (1.f / SCALEF);
  u.w = fmaxf(v.w * inv, 0.f) * (1.f / SCALEF);

  float n2 = u.x * u.x + u.y * u.y + u.z * u.z + u.w * u.w;
  n2 = wave_reduce_add(n2);
  n2 = __shfl(n2, 0, 32);

  const float norm = fmaxf(sqrtf(n2), EPSF);   // c = 1 -> sqrt_c = 1
  const float t = tanhf(norm) / norm;

  float4 o = make_float4(t * u.x, t * u.y, t * u.z, t * u.w);
  ((float4*)(out + (size_t)node * DOUT))[lane] = o;
}

// ---------------- Host launcher ----------------
extern "C" void kernel_launch(void* const* d_in, const int* in_sizes, int n_in,
                              void* d_out, int out_size, void* d_ws, size_t ws_size,
                              hipStream_t stream) {
  const float* x    = (const float*)d_in[0];
  const float* W    = (const float*)d_in[1];
  const float* a    = (const float*)d_in[2];
  const int*   edge = (const int*)d_in[3];

  const int n_nodes = in_sizes[0] / DIN;
  const int n_edges = in_sizes[3] / 2;

  float* out = (float*)d_out;

  // Workspace layout: h [n*128] | s1 [n] | s2 [n] | rowsum [n]  (~26.2 MB)
  float* h      = (float*)d_ws;
  float* s1     = h  + (size_t)n_nodes * DOUT;
  float* s2     = s1 + n_nodes;
  float* rowsum = s2 + n_nodes;

  const int total4 = n_nodes * (DOUT / 4);
  k_zero<<<(total4 + 255) / 256, 256, 0, stream>>>(out, rowsum, n_nodes);

  const int mtiles = (n_nodes + 15) / 16;
  k_gemm<<<mtiles, 256, 0, stream>>>(x, W, h, n_nodes);

  const int node_blocks = (n_nodes + 7) / 8;   // 8 waves per 256-thread block
  k_s1s2<<<node_blocks, 256, 0, stream>>>(h, a, s1, s2, n_nodes);

  const int edge_blocks = (n_edges + 7) / 8;
  k_edge<<<edge_blocks, 256, 0, stream>>>(edge, edge + n_edges, h, s1, s2,
                                          out, rowsum, n_edges);

  k_final<<<node_blocks, 256, 0, stream>>>(out, rowsum, n_nodes);
}